// BatchMatMulModule_11879879541522
// MI455X (gfx1250) — compile-verified
//
#include <hip/hip_runtime.h>

typedef float v2f __attribute__((ext_vector_type(2)));
typedef float v8f __attribute__((ext_vector_type(8)));
typedef float f4  __attribute__((ext_vector_type(4)));

#define NBATCH 4
#define N1     64            // 3x3 matrices per batch
#define MDIM   100000
#define MT     16            // m per tile (one WMMA column... row block)
#define TILES_PER_B (MDIM / MT)      // 6250
#define WAVES_PER_BLOCK 4
#define NROWS  192           // 64*3 flattened (n,i) rows
#define TILE_DW (NROWS * MT) // 3072 dwords staged per wave (12 KB)

// out[b][n][m][i] = sum_j x[b][n][i][j] * y[b][m][j]
// Per (b, m-tile): D = A(16x4: y) @ B(4x16: x rows), 12 chunks over r = 3n+i.
__global__ __launch_bounds__(128) void bmm3_wmma_kernel(
    const float* __restrict__ xf,   // [4][64][3][3]
    const float* __restrict__ yf,   // [4][100000][3]
    float* __restrict__ out)        // [4][64][100000][3]
{
    __shared__ __attribute__((aligned(16))) float lds[WAVES_PER_BLOCK * TILE_DW];

    const int lane = threadIdx.x & 31;
    const int wid  = threadIdx.x >> 5;
    const int gw   = blockIdx.x * WAVES_PER_BLOCK + wid;   // 0..24999
    const int b    = gw / TILES_PER_B;
    const int t    = gw - b * TILES_PER_B;
    const int m0   = t * MT;

    float* w = lds + wid * TILE_DW;

    const int col = lane & 15;       // N index / M index within half
    const int h   = lane >> 4;       // lane half

    // ---- A matrix: 16x4 f32, A[M][K] = y[b][m0+M][K], K=3 zero pad ----
    // Layout (ISA 7.12.2): lanes 0-15: V0=K0, V1=K1 ; lanes 16-31: V0=K2, V1=K3
    const float* yb = yf + (size_t)b * (MDIM * 3);
    v2f a;
    {
        const int m = m0 + col;
        a.x = yb[m * 3 + 2 * h];                    // K = 2h   (j=0 / j=2)
        a.y = (h == 0) ? yb[m * 3 + 1] : 0.0f;      // K = 2h+1 (j=1 / pad)
    }

    // ---- B matrices: 4x16 f32 per chunk, B[K][N] = x_rows[c*16+N][K] ----
    // x_rows[r][j] = x[b][r/3][r%3][j] at xb[r*3 + j]; K=3 zero pad.
    const float* xb = xf + b * (N1 * 9);
    v2f bm[12];
#pragma unroll
    for (int c = 0; c < 12; ++c) {
        const int r = c * 16 + col;
        bm[c].x = xb[r * 3 + 2 * h];                // K = 2h
        bm[c].y = (h == 0) ? xb[r * 3 + 1] : 0.0f;  // K = 2h+1
    }

    // ---- 12 WMMAs; scatter D into LDS in output-linear order [n][m][i] ----
#pragma unroll
    for (int c = 0; c < 12; ++c) {
        v8f acc = {};
        acc = __builtin_amdgcn_wmma_f32_16x16x4_f32(
            /*neg_a=*/false, a, /*neg_b=*/false, bm[c],
            /*c_mod=*/(short)0, acc, /*reuse_a=*/false, /*reuse_b=*/false);

        const int r  = c * 16 + col;     // this lane's D column = row r of x_rows
        const int nr = r / 3;
        const int ir = r - 3 * nr;
        const int offBase = nr * 48 + ir;          // [n][m][i] tile layout
#pragma unroll
        for (int v = 0; v < 8; ++v) {
            const int mloc = v + 8 * h;            // D row M = v + 8*half
            w[offBase + mloc * 3] = acc[v];
        }
    }

    // ---- readback: 64 n-spans of 48 contiguous floats -> NT b128 stores ----
    // LDS ops of one wave are in-order (DScnt), no barrier needed.
    float* ob = out + (size_t)b * ((size_t)N1 * MDIM * 3) + (size_t)m0 * 3;
    for (int p = 0; p < 24; ++p) {
        const int f = p * 32 + lane;               // 0..767 float4 slots
        const int n = f / 12;                      // 0..63
        const int q = f - n * 12;                  // 0..11 float4 within span
        const f4 vv = *(const f4*)(w + n * 48 + q * 4);
        f4* dst = (f4*)(ob + (size_t)n * (MDIM * 3) + q * 4);
        __builtin_nontemporal_store(vv, dst);
    }
}

extern "C" void kernel_launch(void* const* d_in, const int* in_sizes, int n_in,
                              void* d_out, int out_size, void* d_ws, size_t ws_size,
                              hipStream_t stream) {
    const float* x = (const float*)d_in[0];
    const float* y = (const float*)d_in[1];
    float* out = (float*)d_out;
    (void)in_sizes; (void)n_in; (void)out_size; (void)d_ws; (void)ws_size;

    const int total_waves = NBATCH * TILES_PER_B;              // 25000
    const int blocks = total_waves / WAVES_PER_BLOCK;          // 6250
    bmm3_wmma_kernel<<<dim3(blocks), dim3(128), 0, stream>>>(x, y, out);
}